// Frequency_TopK_52785148067884
// MI455X (gfx1250) — compile-verified
//
#include <hip/hip_runtime.h>
#include <math.h>

// Problem constants (fixed by the reference)
#define BB    32
#define CC    768
#define HH    24
#define NTOK  577          // 1 cls + 576 spatial
#define NPIX  576
#define GCH   4            // channels per block
#define NBLK_C (CC/GCH)    // 192 channel-groups
#define RROWS (GCH*HH)     // 96 GEMM rows per block
#define LSTR  25           // padded LDS row stride

typedef __attribute__((ext_vector_type(2))) float v2f;
typedef __attribute__((ext_vector_type(8))) float v8f;

static __device__ __forceinline__ v8f wmma4(v2f a, v2f b, v8f c) {
  // D(16x16,f32) = A(16x4,f32) * B(4x16,f32) + C
  return __builtin_amdgcn_wmma_f32_16x16x4_f32(false, a, false, b, (short)0, c,
                                               false, false);
}

#define V8ZERO {0.f,0.f,0.f,0.f,0.f,0.f,0.f,0.f}

// ---------------------------------------------------------------------------
// Kernel 1: build unitary DFT tables F[j][k] = exp(-2*pi*i*j*k/24)/sqrt(24)
//           and zero the per-batch spectrum accumulator (replay-safe).
// ---------------------------------------------------------------------------
__global__ void fas_init(float* __restrict__ Fre, float* __restrict__ Fim,
                         float* __restrict__ flat_s) {
  int t = blockIdx.x * 256 + threadIdx.x;
  if (t < HH * HH) {
    int j = t / HH, k = t % HH;
    int m = (j * k) % HH;                 // exploit periodicity for accuracy
    float ang = -2.0f * 3.14159265358979323846f * (float)m / (float)HH;
    const float scale = 0.20412414523193154f;   // 1/sqrt(24)
    Fre[t] = cosf(ang) * scale;
    Fim[t] = sinf(ang) * scale;
  }
  if (t < BB * NPIX) flat_s[t] = 0.f;
}

// ---------------------------------------------------------------------------
// Kernel 2: forward FFT2 per (b, channel-group), accumulate shifted amplitude.
// ---------------------------------------------------------------------------
__global__ __launch_bounds__(128) void fas_fft_amp(
    const float* __restrict__ x, const float* __restrict__ gFre,
    const float* __restrict__ gFim, float* __restrict__ flat_s) {
  __shared__ float As[RROWS][LSTR];   // X rows (c,h), cols w
  __shared__ float Bre[RROWS][LSTR];  // T^T rows (c,kw), cols h
  __shared__ float Bim[RROWS][LSTR];
  __shared__ float Fr[HH][32];        // DFT table, cols 24..31 zero-padded
  __shared__ float Fi[HH][32];
  __shared__ float amps[NPIX];

  const int b    = blockIdx.x / NBLK_C;
  const int c0   = (blockIdx.x % NBLK_C) * GCH;
  const int tid  = threadIdx.x;
  const int wave = tid >> 5, lane = tid & 31;
  const int lm   = lane & 15, lh = lane >> 4;   // lane halves for frags

  for (int i = tid; i < HH * 32; i += 128) {
    int r = i >> 5, c = i & 31;
    Fr[r][c] = (c < HH) ? gFre[r * HH + c] : 0.f;
    Fi[r][c] = (c < HH) ? gFim[r * HH + c] : 0.f;
  }
  for (int i = tid; i < NPIX; i += 128) amps[i] = 0.f;
  for (int e = tid; e < RROWS * HH; e += 128) {
    int row = e / HH, w = e % HH;
    int c = row / HH, h = row % HH;
    As[row][w] = x[((size_t)b * NTOK + 1 + h * HH + w) * CC + (c0 + c)];
  }
  __syncthreads();

  // ---- Stage 1: T = X * F  (real * complex); store transposed ----
  for (int j = wave; j < 12; j += 4) {
    const int mt = j >> 1, nt = j & 1;
    const int mrow = mt * 16 + lm;
    const int ncol = nt * 16 + lm;
    v8f ar = V8ZERO, ai = V8ZERO;
    for (int ks = 0; ks < 6; ++ks) {
      const int k0 = ks * 4 + lh * 2;
      v2f a;  a.x  = As[mrow][k0];   a.y  = As[mrow][k0 + 1];
      v2f br; br.x = Fr[k0][ncol];   br.y = Fr[k0 + 1][ncol];
      v2f bi; bi.x = Fi[k0][ncol];   bi.y = Fi[k0 + 1][ncol];
      ar = wmma4(a, br, ar);
      ai = wmma4(a, bi, ai);
    }
    if (ncol < HH) {
      #pragma unroll
      for (int v = 0; v < 8; ++v) {
        int m = mt * 16 + lh * 8 + v;          // (c,h)
        int c = m / HH, h = m % HH;
        Bre[c * HH + ncol][h] = ar[v];
        Bim[c * HH + ncol][h] = ai[v];
      }
    }
  }
  __syncthreads();

  // ---- Stage 2: Y' = T' * F (complex * complex); amplitude accumulate ----
  for (int j = wave; j < 12; j += 4) {
    const int mt = j >> 1, nt = j & 1;
    const int mrow = mt * 16 + lm;
    const int ncol = nt * 16 + lm;
    v8f pr = V8ZERO, qr = V8ZERO, yi = V8ZERO;
    for (int ks = 0; ks < 6; ++ks) {
      const int k0 = ks * 4 + lh * 2;
      v2f are; are.x = Bre[mrow][k0]; are.y = Bre[mrow][k0 + 1];
      v2f aim; aim.x = Bim[mrow][k0]; aim.y = Bim[mrow][k0 + 1];
      v2f br;  br.x  = Fr[k0][ncol];  br.y  = Fr[k0 + 1][ncol];
      v2f bi;  bi.x  = Fi[k0][ncol];  bi.y  = Fi[k0 + 1][ncol];
      pr = wmma4(are, br, pr);      // Tre*Fr
      qr = wmma4(aim, bi, qr);      // Tim*Fi   (Yre = pr - qr)
      yi = wmma4(are, bi, yi);      // Tre*Fi
      yi = wmma4(aim, br, yi);      // + Tim*Fr (Yim)
    }
    if (ncol < HH) {
      #pragma unroll
      for (int v = 0; v < 8; ++v) {
        int m  = mt * 16 + lh * 8 + v;         // (c,kw)
        int kw = m % HH;
        int kh = ncol;
        float re = pr[v] - qr[v];
        float im = yi[v];
        float amp = sqrtf(re * re + im * im);
        int si = ((kh + 12) % HH) * HH + ((kw + 12) % HH);  // fftshift
        atomicAdd(&amps[si], amp);             // ds_add_f32
      }
    }
  }
  __syncthreads();
  for (int p = tid; p < NPIX; p += 128)
    atomicAdd(&flat_s[b * NPIX + p], amps[p]);
}

// ---------------------------------------------------------------------------
// Kernel 3: per-batch, per-radial-band rank-select top-k -> 0/1 mask.
// Tie-breaking (equal value -> lower index wins) matches jax.lax.top_k.
// ---------------------------------------------------------------------------
__global__ __launch_bounds__(NPIX) void fas_mask(
    const float* __restrict__ flat_s, const float* __restrict__ ratio_p,
    float* __restrict__ maskg) {
  __shared__ float vals[NPIX];
  __shared__ int   bands[NPIX];
  const int b = blockIdx.x;
  const int p = threadIdx.x;

  float ratio = ratio_p[0];
  if (!(ratio > 0.f && ratio <= 1.f))          // hedge: scalar stored as f64
    ratio = (float)(((const double*)ratio_p)[0]);

  vals[p] = flat_s[b * NPIX + p];
  int py = p / HH, px = p % HH;
  float dx = (float)(px - 12), dy = (float)(py - 12);
  int ring = (int)floorf(sqrtf(dx * dx + dy * dy));
  bands[p] = (ring <= 4) ? 0 : ((ring <= 8) ? 1 : 2);
  __syncthreads();

  const int bd = bands[p];
  const float v = vals[p];
  int n = 0, rank = 0;
  for (int q = 0; q < NPIX; ++q) {
    if (bands[q] == bd) {
      ++n;
      float u = vals[q];
      if (u > v || (u == v && q < p)) ++rank;
    }
  }
  int k = (int)((float)n * ratio);
  if (k < 1) k = 1;
  maskg[b * NPIX + p] = (rank < k) ? 1.f : 0.f;
}

// ---------------------------------------------------------------------------
// Kernel 4: recompute FFT2, apply mask, inverse FFT2 (real part), write out.
// ---------------------------------------------------------------------------
__global__ __launch_bounds__(128) void fas_filter(
    const float* __restrict__ x, const float* __restrict__ gFre,
    const float* __restrict__ gFim, const float* __restrict__ maskg,
    float* __restrict__ out) {
  __shared__ float As[RROWS][LSTR];
  __shared__ float Bre[RROWS][LSTR];
  __shared__ float Bim[RROWS][LSTR];
  __shared__ float Cre[RROWS][LSTR];  // masked spectrum Z' rows (c,kw), cols kh
  __shared__ float Cim[RROWS][LSTR];
  __shared__ float Fr[HH][32];
  __shared__ float Fi[HH][32];
  __shared__ float msk[NPIX];

  const int b    = blockIdx.x / NBLK_C;
  const int c0   = (blockIdx.x % NBLK_C) * GCH;
  const int tid  = threadIdx.x;
  const int wave = tid >> 5, lane = tid & 31;
  const int lm   = lane & 15, lh = lane >> 4;

  for (int i = tid; i < HH * 32; i += 128) {
    int r = i >> 5, c = i & 31;
    Fr[r][c] = (c < HH) ? gFre[r * HH + c] : 0.f;
    Fi[r][c] = (c < HH) ? gFim[r * HH + c] : 0.f;
  }
  for (int i = tid; i < NPIX; i += 128) msk[i] = maskg[b * NPIX + i];
  for (int e = tid; e < RROWS * HH; e += 128) {
    int row = e / HH, w = e % HH;
    int c = row / HH, h = row % HH;
    As[row][w] = x[((size_t)b * NTOK + 1 + h * HH + w) * CC + (c0 + c)];
  }
  __syncthreads();

  // ---- Stage 1: T = X * F ; store transposed ----
  for (int j = wave; j < 12; j += 4) {
    const int mt = j >> 1, nt = j & 1;
    const int mrow = mt * 16 + lm;
    const int ncol = nt * 16 + lm;
    v8f ar = V8ZERO, ai = V8ZERO;
    for (int ks = 0; ks < 6; ++ks) {
      const int k0 = ks * 4 + lh * 2;
      v2f a;  a.x  = As[mrow][k0];   a.y  = As[mrow][k0 + 1];
      v2f br; br.x = Fr[k0][ncol];   br.y = Fr[k0 + 1][ncol];
      v2f bi; bi.x = Fi[k0][ncol];   bi.y = Fi[k0 + 1][ncol];
      ar = wmma4(a, br, ar);
      ai = wmma4(a, bi, ai);
    }
    if (ncol < HH) {
      #pragma unroll
      for (int v = 0; v < 8; ++v) {
        int m = mt * 16 + lh * 8 + v;
        int c = m / HH, h = m % HH;
        Bre[c * HH + ncol][h] = ar[v];
        Bim[c * HH + ncol][h] = ai[v];
      }
    }
  }
  __syncthreads();

  // ---- Stage 2: Y' = T' * F ; mask; store Z' directly ----
  for (int j = wave; j < 12; j += 4) {
    const int mt = j >> 1, nt = j & 1;
    const int mrow = mt * 16 + lm;
    const int ncol = nt * 16 + lm;
    v8f pr = V8ZERO, qr = V8ZERO, yi = V8ZERO;
    for (int ks = 0; ks < 6; ++ks) {
      const int k0 = ks * 4 + lh * 2;
      v2f are; are.x = Bre[mrow][k0]; are.y = Bre[mrow][k0 + 1];
      v2f aim; aim.x = Bim[mrow][k0]; aim.y = Bim[mrow][k0 + 1];
      v2f br;  br.x  = Fr[k0][ncol];  br.y  = Fr[k0 + 1][ncol];
      v2f bi;  bi.x  = Fi[k0][ncol];  bi.y  = Fi[k0 + 1][ncol];
      pr = wmma4(are, br, pr);
      qr = wmma4(aim, bi, qr);
      yi = wmma4(are, bi, yi);
      yi = wmma4(aim, br, yi);
    }
    if (ncol < HH) {
      #pragma unroll
      for (int v = 0; v < 8; ++v) {
        int m  = mt * 16 + lh * 8 + v;         // (c,kw)
        int kw = m % HH;
        int kh = ncol;
        float mv = msk[((kh + 12) % HH) * HH + ((kw + 12) % HH)];
        Cre[m][ncol] = (pr[v] - qr[v]) * mv;
        Cim[m][ncol] = yi[v] * mv;
      }
    }
  }
  __syncthreads();

  // ---- Stage 3: V = Z' * conj(F) ; store transposed (rows (c,h), cols kw) ----
  for (int j = wave; j < 12; j += 4) {
    const int mt = j >> 1, nt = j & 1;
    const int mrow = mt * 16 + lm;
    const int ncol = nt * 16 + lm;
    v8f r1 = V8ZERO, r2 = V8ZERO, i1 = V8ZERO, i2 = V8ZERO;
    for (int ks = 0; ks < 6; ++ks) {
      const int k0 = ks * 4 + lh * 2;
      v2f zre; zre.x = Cre[mrow][k0]; zre.y = Cre[mrow][k0 + 1];
      v2f zim; zim.x = Cim[mrow][k0]; zim.y = Cim[mrow][k0 + 1];
      v2f br;  br.x  = Fr[k0][ncol];  br.y  = Fr[k0 + 1][ncol];
      v2f bi;  bi.x  = Fi[k0][ncol];  bi.y  = Fi[k0 + 1][ncol];
      r1 = wmma4(zre, br, r1);      // Zre*Fr
      r2 = wmma4(zim, bi, r2);      // Zim*Fi   (Vre = r1 + r2)
      i1 = wmma4(zim, br, i1);      // Zim*Fr
      i2 = wmma4(zre, bi, i2);      // Zre*Fi   (Vim = i1 - i2)
    }
    if (ncol < HH) {
      #pragma unroll
      for (int v = 0; v < 8; ++v) {
        int m  = mt * 16 + lh * 8 + v;         // (c,kw)
        int c  = m / HH, kw = m % HH;
        Bre[c * HH + ncol][kw] = r1[v] + r2[v];
        Bim[c * HH + ncol][kw] = i1[v] - i2[v];
      }
    }
  }
  __syncthreads();

  // ---- Stage 4: O = Re( V' * conj(F) ) ; write to output tokens ----
  for (int j = wave; j < 12; j += 4) {
    const int mt = j >> 1, nt = j & 1;
    const int mrow = mt * 16 + lm;
    const int ncol = nt * 16 + lm;
    v8f o1 = V8ZERO, o2 = V8ZERO;
    for (int ks = 0; ks < 6; ++ks) {
      const int k0 = ks * 4 + lh * 2;
      v2f vre; vre.x = Bre[mrow][k0]; vre.y = Bre[mrow][k0 + 1];
      v2f vim; vim.x = Bim[mrow][k0]; vim.y = Bim[mrow][k0 + 1];
      v2f br;  br.x  = Fr[k0][ncol];  br.y  = Fr[k0 + 1][ncol];
      v2f bi;  bi.x  = Fi[k0][ncol];  bi.y  = Fi[k0 + 1][ncol];
      o1 = wmma4(vre, br, o1);      // Vre*Fr
      o2 = wmma4(vim, bi, o2);      // Vim*Fi  (real part = o1 + o2)
    }
    if (ncol < HH) {
      #pragma unroll
      for (int v = 0; v < 8; ++v) {
        int m = mt * 16 + lh * 8 + v;          // (c,h)
        int c = m / HH, h = m % HH;
        out[((size_t)b * NTOK + 1 + h * HH + ncol) * CC + (c0 + c)] =
            o1[v] + o2[v];
      }
    }
  }
}

// ---------------------------------------------------------------------------
// Kernel 5: copy the class token row untouched.
// ---------------------------------------------------------------------------
__global__ void fas_cls(const float* __restrict__ x, float* __restrict__ out) {
  int t = blockIdx.x * 256 + threadIdx.x;
  if (t < BB * CC) {
    int b = t / CC, c = t % CC;
    out[(size_t)b * NTOK * CC + c] = x[(size_t)b * NTOK * CC + c];
  }
}

// ---------------------------------------------------------------------------
extern "C" void kernel_launch(void* const* d_in, const int* in_sizes, int n_in,
                              void* d_out, int out_size, void* d_ws,
                              size_t ws_size, hipStream_t stream) {
  (void)in_sizes; (void)n_in; (void)out_size; (void)ws_size;
  const float* x     = (const float*)d_in[0];
  const float* ratio = (const float*)d_in[1];
  float* out = (float*)d_out;

  float* ws    = (float*)d_ws;
  float* Fre   = ws;                          // 576
  float* Fim   = ws + 576;                    // 576
  float* flat  = ws + 1152;                   // 32*576
  float* maskg = ws + 1152 + BB * NPIX;       // 32*576

  fas_init<<<(BB * NPIX + 255) / 256, 256, 0, stream>>>(Fre, Fim, flat);
  fas_cls<<<(BB * CC + 255) / 256, 256, 0, stream>>>(x, out);
  fas_fft_amp<<<BB * NBLK_C, 128, 0, stream>>>(x, Fre, Fim, flat);
  fas_mask<<<BB, NPIX, 0, stream>>>(flat, ratio, maskg);
  fas_filter<<<BB * NBLK_C, 128, 0, stream>>>(x, Fre, Fim, maskg, out);
}